// QbfAttention_24507083391179
// MI455X (gfx1250) — compile-verified
//
#include <hip/hip_runtime.h>
#include <hip/hip_bf16.h>
#include <stdint.h>

typedef float v2f __attribute__((ext_vector_type(2)));
typedef float v8f __attribute__((ext_vector_type(8)));

#define BS 32
#define VARNUM 16384
#define EMB 128
#define NHEADS 16
#define WSPL 64               // softmax splits per batch (1 wave each)
#define ROWS_PER_WAVE (VARNUM / WSPL)       // 256
#define TILES_PER_WAVE (ROWS_PER_WAVE / 16) // 16
#define LSTR 132              // padded LDS row stride (floats), kills bank conflicts
#define BIGNEG (-3.0e38f)

// ---------------------------------------------------------------------------
// Kernel 1: qs = tanh(W_q @ seed), L2 normalize per head, gram loss partial.
// One block per batch.
// ---------------------------------------------------------------------------
__global__ void __launch_bounds__(256) k1_qs(
    const float* __restrict__ seed, const float* __restrict__ wqs,
    float* __restrict__ normq, float* __restrict__ lossp) {
  const int b = blockIdx.x, t = threadIdx.x;
  __shared__ __align__(16) float s_seed[128];
  __shared__ __align__(16) float s_q[NHEADS * EMB];
  __shared__ float s_inv[NHEADS];
  __shared__ float s_red[8];

  if (t < 128) s_seed[t] = seed[b * 128 + t];
  __syncthreads();

  float vals[8];
#pragma unroll
  for (int i = 0; i < 8; ++i) {
    int idx = t + 256 * i;                       // idx = h*128 + e
    const float4* wr = (const float4*)(wqs + (size_t)idx * 128);
    float acc = 0.f;
#pragma unroll 8
    for (int s4 = 0; s4 < 32; ++s4) {
      float4 w = wr[s4];
      float4 sd = *(const float4*)&s_seed[s4 * 4];
      acc += w.x * sd.x + w.y * sd.y + w.z * sd.z + w.w * sd.w;
    }
    vals[i] = tanhf(acc);
    s_q[idx] = vals[i];
  }
  __syncthreads();

  {  // per-head inverse L2 norm: 16 threads/head (threads grouped by t>>4)
    int h = t >> 4, j = t & 15;
    float p = 0.f;
    for (int e = j; e < EMB; e += 16) { float q = s_q[h * EMB + e]; p += q * q; }
    p += __shfl_xor(p, 8, 32); p += __shfl_xor(p, 4, 32);
    p += __shfl_xor(p, 2, 32); p += __shfl_xor(p, 1, 32);
    if (j == 0) s_inv[h] = rsqrtf(p);
  }
  __syncthreads();

#pragma unroll
  for (int i = 0; i < 8; ++i) {
    int idx = t + 256 * i;
    float nv = vals[i] * s_inv[idx >> 7];
    s_q[idx] = nv;
    normq[(size_t)b * (NHEADS * EMB) + idx] = nv;
  }
  __syncthreads();

  {  // gram - I, squared, reduce; 256 threads <-> 256 (i,j) head pairs
    int i = t >> 4, j = t & 15;
    float d = 0.f;
    for (int e = 0; e < EMB; ++e) d += s_q[i * EMB + e] * s_q[j * EMB + e];
    d -= (i == j) ? 1.f : 0.f;
    float p = d * d;
    p += __shfl_xor(p, 16, 32); p += __shfl_xor(p, 8, 32);
    p += __shfl_xor(p, 4, 32);  p += __shfl_xor(p, 2, 32);
    p += __shfl_xor(p, 1, 32);
    if ((t & 31) == 0) s_red[t >> 5] = p;
    __syncthreads();
    if (t == 0) {
      float s = 0.f;
      for (int w = 0; w < 8; ++w) s += s_red[w];
      lossp[b] = s * (1.0f / (float)BS);
    }
  }
}

// ---------------------------------------------------------------------------
// Async DMA of one 16x128 f32 tile (8 KB) into a per-wave LDS buffer.
// 16 x global_load_async_to_lds_b128: no VGPR staging, tracked by ASYNCcnt.
// LDS byte offset = low 32 bits of the generic pointer (flat LDS aperture).
// ---------------------------------------------------------------------------
__device__ __forceinline__ void issue_tile_async(const float4* ep, float* buf,
                                                 int lane) {
#pragma unroll
  for (int i = 0; i < 16; ++i) {
    int idx = lane + i * 32;
    unsigned ldsoff =
        (unsigned)(uintptr_t)&buf[(idx >> 5) * LSTR + (idx & 31) * 4];
    const float4* g = ep + idx;
    asm volatile("global_load_async_to_lds_b128 %0, %1, off"
                 :: "v"(ldsoff), "v"(g)
                 : "memory");
  }
}

// ---------------------------------------------------------------------------
// Kernel 2: fused flash attention over embeddings, fp32 WMMA 16x16x4.
// Block = 128 threads = 4 independent waves; each wave = one softmax split.
// Double-buffered async global->LDS staging.
// ---------------------------------------------------------------------------
__global__ void __launch_bounds__(128) k2_flash(
    const float* __restrict__ emb, const int* __restrict__ mask,
    const float* __restrict__ normq, float* __restrict__ part) {
  __shared__ __align__(16) float NQ[NHEADS * LSTR];
  __shared__ __align__(16) float ET[4][2][16 * LSTR];   // per-wave double buffer

  const int tid = threadIdx.x;
  const int wave = tid >> 5;
  const int lane = tid & 31;
  const int b = blockIdx.x >> 4;                    // 16 blocks per batch
  const int w = ((blockIdx.x & 15) << 2) + wave;    // split id 0..63
  const int n16 = lane & 15;
  const int hi = lane >> 4;

  // normq [16x128] -> LDS (padded stride)
  for (int i = tid; i < NHEADS * EMB; i += 128)
    NQ[(i >> 7) * LSTR + (i & 127)] = normq[(size_t)b * (NHEADS * EMB) + i];
  __syncthreads();

  const float4* ebase =
      (const float4*)(emb + ((size_t)b * VARNUM + (size_t)w * ROWS_PER_WAVE) * EMB);
  const int* mbase = mask + b * (VARNUM / 2);

  v8f acc[8];
#pragma unroll
  for (int ct = 0; ct < 8; ++ct) acc[ct] = (v8f)0.f;
  float m = BIGNEG, lsum = 0.f;

  // prologue: DMA tile 0 into buffer 0
  issue_tile_async(ebase, &ET[wave][0][0], lane);

  for (int tile = 0; tile < TILES_PER_WAVE; ++tile) {
    float* E = &ET[wave][tile & 1][0];

    if (tile + 1 < TILES_PER_WAVE) {
      // prior ds reads of the buffer being refilled must drain first (WAR)
      asm volatile("s_wait_dscnt 0" ::: "memory");
      issue_tile_async(ebase + (tile + 1) * 512, &ET[wave][(tile + 1) & 1][0],
                       lane);
      // async loads complete in order: <=16 outstanding => tile `tile` landed
      asm volatile("s_wait_asynccnt 16" ::: "memory");
    } else {
      asm volatile("s_wait_asynccnt 0" ::: "memory");
    }

    const int v0 = w * ROWS_PER_WAVE + tile * 16;
    // mask bits for the 8 rows this lane-half owns (v = v0 + r + 8*hi)
    const int mo = (v0 + 8 * hi) & (VARNUM / 2 - 1);  // 8-aligned, no wrap in run
    int4 ma = *(const int4*)&mbase[mo];
    int4 mb2 = *(const int4*)&mbase[mo + 4];
    int mk[8] = {ma.x, ma.y, ma.z, ma.w, mb2.x, mb2.y, mb2.z, mb2.w};

    // ---- per-row inverse L2 norm (lane handles row n16, half the columns) --
    float rn;
    {
      float s = 0.f;
      const float* rp = &E[n16 * LSTR + 64 * hi];
#pragma unroll
      for (int i = 0; i < 16; ++i) {
        float4 q = *(const float4*)&rp[4 * i];
        s += q.x * q.x + q.y * q.y + q.z * q.z + q.w * q.w;
      }
      s += __shfl_xor(s, 16, 32);
      rn = rsqrtf(s);
    }

    // ---- logits' [v][h] = E(16x128) x normQ^T(128x16): 32 WMMA K-slices ----
    v8f lg = (v8f)0.f;
    const int aoff = n16 * LSTR + 2 * hi;
#pragma unroll
    for (int c = 0; c < 32; ++c) {
      v2f a = *(const v2f*)&E[aoff + 4 * c];    // A[v][k] slice
      v2f bq = *(const v2f*)&NQ[aoff + 4 * c];  // B[k][h] = Q[h][k] slice
      lg = __builtin_amdgcn_wmma_f32_16x16x4_f32(false, a, false, bq,
                                                 (short)0, lg, false, false);
    }

    // ---- mask + cosine scaling + online softmax (head = n16 per lane) ------
    float p[8];
    float tmax = BIGNEG;
#pragma unroll
    for (int r = 0; r < 8; ++r) {
      float rr = __shfl(rn, r + 8 * hi, 32);    // inv-norm of row r+8*hi
      float lr = (mk[r] != 0) ? lg[r] * rr : BIGNEG;
      p[r] = lr;
      tmax = fmaxf(tmax, lr);
    }
    tmax = fmaxf(tmax, __shfl_xor(tmax, 16, 32));
    float mnew = fmaxf(m, tmax);
    float sc = __expf(m - mnew);
    float tsum = 0.f;
#pragma unroll
    for (int r = 0; r < 8; ++r) {
      float pe = (mk[r] != 0) ? __expf(p[r] - mnew) : 0.f;
      p[r] = pe;
      tsum += pe;
    }
    tsum += __shfl_xor(tsum, 16, 32);
    lsum = lsum * sc + tsum;
    m = mnew;

    // rescale accumulators: row r of acc holds head r+8*hi
#pragma unroll
    for (int r = 0; r < 8; ++r) {
      float scr = __shfl(sc, r + 8 * hi, 32);
#pragma unroll
      for (int ct = 0; ct < 8; ++ct) acc[ct][r] *= scr;
    }

    // ---- build A-layout slices of P (A[h][v]) from D'-layout p[] -----------
    float q0 = __shfl_xor(p[0], 16, 32), q1 = __shfl_xor(p[1], 16, 32);
    float q2 = __shfl_xor(p[2], 16, 32), q3 = __shfl_xor(p[3], 16, 32);
    float q4 = __shfl_xor(p[4], 16, 32), q5 = __shfl_xor(p[5], 16, 32);
    float q6 = __shfl_xor(p[6], 16, 32), q7 = __shfl_xor(p[7], 16, 32);
    v2f aS[4];
    aS[0].x = hi ? q2 : p[0];  aS[0].y = hi ? q3 : p[1];
    aS[1].x = hi ? q6 : p[4];  aS[1].y = hi ? q7 : p[5];
    aS[2].x = hi ? p[2] : q0;  aS[2].y = hi ? p[3] : q1;
    aS[3].x = hi ? p[6] : q4;  aS[3].y = hi ? p[7] : q5;

    // ---- rc += P^T(16x16) x E(16x128): 8 col-tiles x 4 K-slices ------------
#pragma unroll
    for (int ct = 0; ct < 8; ++ct) {
#pragma unroll
      for (int s = 0; s < 4; ++s) {
        v2f bb;
        bb.x = E[(4 * s + 2 * hi + 0) * LSTR + 16 * ct + n16];
        bb.y = E[(4 * s + 2 * hi + 1) * LSTR + 16 * ct + n16];
        acc[ct] = __builtin_amdgcn_wmma_f32_16x16x4_f32(
            false, aS[s], false, bb, (short)0, acc[ct], false, false);
      }
    }
  }

  // ---- write split partial: [b][w][h]{acc[0:128], m, l} ----
  float* pb = part + ((size_t)(b * WSPL + w)) * NHEADS * 130;
#pragma unroll
  for (int r = 0; r < 8; ++r) {
    int h = r + 8 * hi;
#pragma unroll
    for (int ct = 0; ct < 8; ++ct) pb[h * 130 + 16 * ct + n16] = acc[ct][r];
  }
  if (lane < 16) {
    pb[lane * 130 + 128] = m;
    pb[lane * 130 + 129] = lsum;
  }
}

// ---------------------------------------------------------------------------
// Kernel 3: combine split-softmax partials; block per (b,h), 128 threads.
// ---------------------------------------------------------------------------
__global__ void __launch_bounds__(128) k3_combine(
    const float* __restrict__ part, const float* __restrict__ lossp,
    float* __restrict__ out) {
  const int blk = blockIdx.x;
  const int b = blk >> 4, h = blk & 15;
  const int e = threadIdx.x;
  const float* pb = part + ((size_t)b * WSPL) * NHEADS * 130 + h * 130;
  float M = BIGNEG;
  for (int w = 0; w < WSPL; ++w)
    M = fmaxf(M, pb[(size_t)w * NHEADS * 130 + 128]);
  float L = 0.f, A = 0.f;
  for (int w = 0; w < WSPL; ++w) {
    const float* pw = pb + (size_t)w * NHEADS * 130;
    float s = __expf(pw[128] - M);
    L += s * pw[129];
    A += s * pw[e];
  }
  out[((size_t)b * NHEADS + h) * EMB + e] = A / L;
  if (blk == 0 && e == 0) {
    float s = 0.f;
    for (int i = 0; i < BS; ++i) s += lossp[i];
    out[BS * NHEADS * EMB] = s;   // loss slot
  }
}

// ---------------------------------------------------------------------------
extern "C" void kernel_launch(void* const* d_in, const int* in_sizes, int n_in,
                              void* d_out, int out_size, void* d_ws, size_t ws_size,
                              hipStream_t stream) {
  const float* query_seed = (const float*)d_in[0];   // [32,128]
  const float* embeddings = (const float*)d_in[1];   // [32,16384,128]
  const int*   attn_mask  = (const int*)d_in[2];     // [32,8192]
  const float* w_qs       = (const float*)d_in[3];   // [16,128,128]
  float* out = (float*)d_out;                        // rc[32*16*128] + loss[1]

  float* ws    = (float*)d_ws;
  float* normq = ws;                                  // 65536 floats
  float* lossp = ws + 65536;                          // 32 floats
  float* part  = ws + 65600;                          // 32*64*16*130 floats

  k1_qs<<<BS, 256, 0, stream>>>(query_seed, w_qs, normq, lossp);
  k2_flash<<<BS * (WSPL / 4), 128, 0, stream>>>(embeddings, attn_mask, normq, part);
  k3_combine<<<BS * NHEADS, 128, 0, stream>>>(part, lossp, out);
}